// GraphGCN_simple_54614804136601
// MI455X (gfx1250) — compile-verified
//
#include <hip/hip_runtime.h>
#include <hip/hip_bf16.h>

typedef __attribute__((ext_vector_type(2))) float v2f;
typedef __attribute__((ext_vector_type(8))) float v8f;

#define N_FEAT    128
#define EMBED     20
#define N_GRAPHS  64
#define N_CLASSES 10
#define WPAD      (N_FEAT + 2)   // LDS row pad to spread banks

// ---------------- init: deg=1 (self-loop weight), out1=0, pool buffers=0 ----
__global__ void k_init(float* __restrict__ deg, float* __restrict__ out1,
                       float* __restrict__ pool, int N, int poolN) {
  int i = blockIdx.x * blockDim.x + threadIdx.x;
  int stride = gridDim.x * blockDim.x;
  for (int t = i; t < N; t += stride) deg[t] = 1.0f;
  int M = N * EMBED;
  for (int t = i; t < M; t += stride) out1[t] = 0.0f;
  for (int t = i; t < poolN; t += stride) pool[t] = 0.0f;
}

// ---------------- degree: deg[dst] += w ------------------------------------
__global__ void k_degree(const long long* __restrict__ dst,
                         const float* __restrict__ w,
                         float* __restrict__ deg, int E) {
  int i = blockIdx.x * blockDim.x + threadIdx.x;
  if (i < E) atomicAdd(&deg[(int)dst[i]], w[i]);
}

// ---------------- dinv = rsqrt(deg) in place -------------------------------
__global__ void k_dinv(float* __restrict__ deg, int N) {
  int i = blockIdx.x * blockDim.x + threadIdx.x;
  if (i < N) {
    float d = deg[i];
    deg[i] = (d > 0.0f) ? rsqrtf(d) : 0.0f;
  }
}

// ---------------- xw = x @ W via V_WMMA_F32_16X16X4_F32 --------------------
// W staged once per block into LDS, transposed + zero-padded to 32 columns:
// inner loop is branch-free: global_load_b64 (A) + 2x ds_load_b64 (B) + 2 WMMA.
__global__ void k_gemm(const float* __restrict__ x, const float* __restrict__ W,
                       float* __restrict__ xw, int N, int nTiles) {
  __shared__ float Wt[32][WPAD];           // Wt[col][k], cols 20..31 = 0
  int tid = threadIdx.x;
  for (int t = tid; t < 32 * N_FEAT; t += blockDim.x) {
    int col = t >> 7;                      // t / 128
    int k   = t & (N_FEAT - 1);
    Wt[col][k] = (col < EMBED) ? W[k * EMBED + col] : 0.0f;
  }
  __syncthreads();

  int wave = blockIdx.x * (blockDim.x >> 5) + (tid >> 5);
  if (wave >= nTiles) return;              // wave-uniform: EXEC all-ones inside
  int lane = tid & 31;
  int half = lane >> 4;                    // 0: K=k,k+1   1: K=k+2,k+3
  int l    = lane & 15;
  int row0 = wave * 16;
  int arow = row0 + l;
  if (arow >= N) arow = N - 1;             // clamp (N divisible by 16 anyway)
  const float* __restrict__ xr = x + (size_t)arow * N_FEAT;
  const float* __restrict__ b0p = &Wt[l][0];
  const float* __restrict__ b1p = &Wt[16 + l][0];

  v8f acc0 = {};
  v8f acc1 = {};
  #pragma unroll 4
  for (int k = 0; k < N_FEAT; k += 4) {
    int ka = k + 2 * half;                 // even -> 8B-aligned b64 loads
    v2f a;  a.x = xr[ka];                  a.y = xr[ka + 1];
    v2f b0 = *(const v2f*)(b0p + ka);      // ds_load_b64
    v2f b1 = *(const v2f*)(b1p + ka);      // ds_load_b64
    acc0 = __builtin_amdgcn_wmma_f32_16x16x4_f32(false, a, false, b0,
                                                 (short)0, acc0, false, false);
    acc1 = __builtin_amdgcn_wmma_f32_16x16x4_f32(false, a, false, b1,
                                                 (short)0, acc1, false, false);
  }
  // D layout: VGPR r -> row = r + 8*half (within tile), col = l
  int c1 = 16 + l;
  bool c1ok = (c1 < EMBED);
  #pragma unroll
  for (int r = 0; r < 8; r++) {
    int row = row0 + r + 8 * half;
    if (row < N) {
      xw[(size_t)row * EMBED + l] = acc0[r];
      if (c1ok) xw[(size_t)row * EMBED + c1] = acc1[r];
    }
  }
}

// ---------------- scatter: out1[dst] += xw[src] * norm ---------------------
__global__ void k_scatter(const long long* __restrict__ src,
                          const long long* __restrict__ dst,
                          const float* __restrict__ w,
                          const float* __restrict__ dinv,
                          const float* __restrict__ xw,
                          float* __restrict__ out1, int E) {
  int e = blockIdx.x * blockDim.x + threadIdx.x;
  if (e >= E) return;
  int s = (int)src[e];
  int d = (int)dst[e];
  float nrm = dinv[s] * w[e] * dinv[d];
  const float4* __restrict__ xs = reinterpret_cast<const float4*>(xw + (size_t)s * EMBED);
  float* __restrict__ od = out1 + (size_t)d * EMBED;
  #pragma unroll
  for (int q = 0; q < EMBED / 4; q++) {
    float4 v = xs[q];                      // global_load_b128 (rows 16B aligned)
    atomicAdd(&od[q * 4 + 0], v.x * nrm);
    atomicAdd(&od[q * 4 + 1], v.y * nrm);
    atomicAdd(&od[q * 4 + 2], v.z * nrm);
    atomicAdd(&od[q * 4 + 3], v.w * nrm);
  }
}

// ---------------- finalize node: self-loop + bias + L2norm + ReLU, pool ----
__global__ void k_node_pool(const float* __restrict__ xw,
                            const float* __restrict__ dinv,
                            const float* __restrict__ b,
                            const float* __restrict__ out1,
                            const long long* __restrict__ batch,
                            float* __restrict__ mx, float* __restrict__ sum,
                            float* __restrict__ cnt, int N) {
  __shared__ float smx[N_GRAPHS * EMBED];
  __shared__ float ssum[N_GRAPHS * EMBED];
  __shared__ float scnt[N_GRAPHS];
  int tid = threadIdx.x;
  for (int t = tid; t < N_GRAPHS * EMBED; t += blockDim.x) { smx[t] = 0.0f; ssum[t] = 0.0f; }
  for (int t = tid; t < N_GRAPHS; t += blockDim.x) scnt[t] = 0.0f;
  __syncthreads();

  int i = blockIdx.x * blockDim.x + tid;
  if (i < N) {
    float di   = dinv[i];
    float self = di * di;                  // self-loop norm = dinv*1*dinv
    const float4* __restrict__ x4 = reinterpret_cast<const float4*>(xw  + (size_t)i * EMBED);
    const float4* __restrict__ o4 = reinterpret_cast<const float4*>(out1 + (size_t)i * EMBED);
    float v[EMBED];
    float nrm2 = 0.0f;
    #pragma unroll
    for (int q = 0; q < EMBED / 4; q++) {
      float4 xv = x4[q];
      float4 ov = o4[q];
      float* vp = &v[q * 4];
      vp[0] = ov.x + xv.x * self + b[q * 4 + 0];
      vp[1] = ov.y + xv.y * self + b[q * 4 + 1];
      vp[2] = ov.z + xv.z * self + b[q * 4 + 2];
      vp[3] = ov.w + xv.w * self + b[q * 4 + 3];
      nrm2 += vp[0]*vp[0] + vp[1]*vp[1] + vp[2]*vp[2] + vp[3]*vp[3];
    }
    float inv = 1.0f / fmaxf(sqrtf(nrm2), 1e-12f);
    int g = (int)batch[i];
    #pragma unroll
    for (int j = 0; j < EMBED; j++) {
      float e = fmaxf(v[j] * inv, 0.0f);   // ReLU -> e >= 0
      // nonneg float bits are monotonic as uint -> integer atomicMax is valid
      atomicMax(reinterpret_cast<unsigned int*>(&smx[g * EMBED + j]), __float_as_uint(e));
      atomicAdd(&ssum[g * EMBED + j], e);
    }
    atomicAdd(&scnt[g], 1.0f);
  }
  __syncthreads();

  for (int t = tid; t < N_GRAPHS * EMBED; t += blockDim.x) {
    float m = smx[t];
    if (m > 0.0f)
      atomicMax(reinterpret_cast<unsigned int*>(&mx[t]), __float_as_uint(m));
    float s = ssum[t];
    if (s != 0.0f) atomicAdd(&sum[t], s);
  }
  for (int t = tid; t < N_GRAPHS; t += blockDim.x) {
    float c = scnt[t];
    if (c != 0.0f) atomicAdd(&cnt[t], c);
  }
}

// ---------------- final linear: out = [mx, mean] @ lin_W + lin_b -----------
__global__ void k_final(const float* __restrict__ mx, const float* __restrict__ sum,
                        const float* __restrict__ cnt,
                        const float* __restrict__ lin_W, const float* __restrict__ lin_b,
                        float* __restrict__ out) {
  int g = blockIdx.x;
  int c = threadIdx.x;
  if (c >= N_CLASSES) return;
  float cn  = fmaxf(cnt[g], 1.0f);
  float acc = lin_b[c];
  #pragma unroll
  for (int k = 0; k < EMBED; k++)
    acc += mx[g * EMBED + k] * lin_W[k * N_CLASSES + c];
  #pragma unroll
  for (int k = 0; k < EMBED; k++)
    acc += (sum[g * EMBED + k] / cn) * lin_W[(EMBED + k) * N_CLASSES + c];
  out[g * N_CLASSES + c] = acc;
}

extern "C" void kernel_launch(void* const* d_in, const int* in_sizes, int n_in,
                              void* d_out, int out_size, void* d_ws, size_t ws_size,
                              hipStream_t stream) {
  const float*     x      = (const float*)d_in[0];
  const long long* eidx   = (const long long*)d_in[1];
  const float*     ew     = (const float*)d_in[2];
  const long long* batch  = (const long long*)d_in[3];
  const float*     W      = (const float*)d_in[4];
  const float*     b      = (const float*)d_in[5];
  const float*     lin_W  = (const float*)d_in[6];
  const float*     lin_b  = (const float*)d_in[7];
  float*           out    = (float*)d_out;

  const int N = in_sizes[0] / N_FEAT;
  const int E = in_sizes[1] / 2;
  const long long* src = eidx;
  const long long* dst = eidx + E;

  // ---- workspace carve-up (16B aligned blocks) ----
  char* ws = (char*)d_ws;
  size_t off = 0;
  float* deg = (float*)(ws + off);                 // N floats (becomes dinv)
  off += ((size_t)N * 4 + 15) & ~(size_t)15;
  float* xw = (float*)(ws + off);                  // N*EMBED
  off += ((size_t)N * EMBED * 4 + 15) & ~(size_t)15;
  float* out1 = (float*)(ws + off);                // N*EMBED
  off += ((size_t)N * EMBED * 4 + 15) & ~(size_t)15;
  float* mx = (float*)(ws + off);                  // G*EMBED
  off += (size_t)N_GRAPHS * EMBED * 4;
  float* sum = (float*)(ws + off);                 // G*EMBED
  off += (size_t)N_GRAPHS * EMBED * 4;
  float* cnt = (float*)(ws + off);                 // G
  off += (size_t)N_GRAPHS * 4;
  (void)ws_size;

  const int poolN = N_GRAPHS * EMBED * 2 + N_GRAPHS;  // mx+sum+cnt contiguous

  // 1) init
  k_init<<<1024, 256, 0, stream>>>(deg, out1, mx, N, poolN);
  // 2) degree
  k_degree<<<(E + 255) / 256, 256, 0, stream>>>(dst, ew, deg, E);
  // 3) dinv
  k_dinv<<<(N + 255) / 256, 256, 0, stream>>>(deg, N);
  // 4) xw = x @ W  (WMMA, one wave per 16-row tile, 8 waves/block)
  {
    int nTiles = (N + 15) / 16;
    int blocks = (nTiles + 7) / 8;
    k_gemm<<<blocks, 256, 0, stream>>>(x, W, xw, N, nTiles);
  }
  // 5) edge scatter
  k_scatter<<<(E + 255) / 256, 256, 0, stream>>>(src, dst, ew, deg, xw, out1, E);
  // 6) node finalize + LDS-staged pooling
  k_node_pool<<<(N + 255) / 256, 256, 0, stream>>>(xw, deg, b, out1, batch, mx, sum, cnt, N);
  // 7) final linear
  k_final<<<N_GRAPHS, 32, 0, stream>>>(mx, sum, cnt, lin_W, lin_b, out);
  (void)out_size; (void)n_in;
}